// AnchorTargetLayer_69810398429610
// MI455X (gfx1250) — compile-verified
//
#include <hip/hip_runtime.h>
#include <hip/hip_bf16.h>

// ---------------- problem constants ----------------
#define N_IMG   8
#define KA      9
#define HH      120
#define WW      120
#define HWC     (HH*WW)          // 14400 cells
#define A_TOT   (KA*HWC)         // 129600 anchors per image
#define M_GT    64
#define IMG_SZ  1920.0f
#define UPPER_T 0.4f
#define LOWER_T 0.1f
#define NPOSMAX 128
#define NNEGMAX 60
#define BETA    0.1f
#define EPSV    1e-6f
#define TILE_C  256              // cells per block in match_kernel

// ---- CDNA5 feature detection (device pass only; host pass sees fallbacks) ----
#if defined(__gfx1250__) && __has_builtin(__builtin_amdgcn_tensor_load_to_lds) && \
    __has_builtin(__builtin_amdgcn_s_wait_tensorcnt)
#define HAVE_TDM 1
#else
#define HAVE_TDM 0
#endif
#if defined(__gfx1250__) && __has_builtin(__builtin_amdgcn_global_load_async_to_lds_b128)
#define HAVE_ASYNC 1
#else
#define HAVE_ASYNC 0
#endif

#define AS1 __attribute__((address_space(1)))
#define AS3 __attribute__((address_space(3)))

typedef __attribute__((ext_vector_type(4)))  unsigned v4u;
typedef __attribute__((ext_vector_type(4)))  int      v4i;
typedef __attribute__((ext_vector_type(8)))  int      v8i;
typedef __attribute__((ext_vector_type(2)))  float    v2f;
typedef __attribute__((ext_vector_type(8)))  float    v8f;

#if defined(__gfx1250__)
typedef AS1 v4i* gvp4;   // global int32x4*  (async b128 source)
typedef AS3 v4i* lvp4;   // LDS    int32x4*  (async b128 destination)
#endif

// anchor widths/heights: k = ratio*3 + scale; stride=16, scales {8,16,32}, ratios {0.5,1,2}
__device__ __constant__ float c_ws[9] = {
    181.01933598f, 362.03867197f, 724.07734394f,
    128.0f,        256.0f,        512.0f,
     90.50966799f, 181.01933598f, 362.03867197f};
__device__ __constant__ float c_hs[9] = {
     90.50966799f, 181.01933598f, 362.03867197f,
    128.0f,        256.0f,        512.0f,
    181.01933598f, 362.03867197f, 724.07734394f};

__device__ __forceinline__ float clip1920(float v) {
    return fminf(fmaxf(v, 0.0f), IMG_SZ);
}
__device__ __forceinline__ float softplusf(float x) {
    float ax = fabsf(x);
    return fmaxf(x, 0.0f) + log1pf(__expf(-ax));
}
__device__ __forceinline__ float smoothl1(float d) {
    float ad = fabsf(d);
    return (ad < BETA) ? (0.5f * d * d / BETA) : (ad - 0.5f * BETA);
}
__device__ __forceinline__ unsigned ballot32(bool p) {
#if __has_builtin(__builtin_amdgcn_ballot_w32)
    return __builtin_amdgcn_ballot_w32(p);
#else
    return (unsigned)__ballot(p);
#endif
}
__device__ __forceinline__ void wait_async0() {
#if __has_builtin(__builtin_amdgcn_s_wait_asynccnt)
    __builtin_amdgcn_s_wait_asynccnt(0);
#else
    asm volatile("s_wait_asynccnt 0x0" ::: "memory");
#endif
}

// ---------------------------------------------------------------------------
// Kernel 1: per-anchor best IoU / argmax vs 64 GT boxes -> match code.
//   match >= 0: positive (gt index); -1: negative; -2: ignore.
// Data movement (CDNA5 paths):
//   * delta tile (36 ch x 256 cells, row stride H*W) -> LDS via TDM
//     tensor_load_to_lds (2D D#), waited with s_wait_tensorcnt.
//   * GT boxes (64 x float4) -> LDS via global_load_async_to_lds_b128
//     (ASYNCcnt), so the VALU stays free for the IoU inner loop.
// ---------------------------------------------------------------------------
__global__ void __launch_bounds__(256)
match_kernel(const float* __restrict__ deltas, const float* __restrict__ gt,
             int* __restrict__ matches) {
    const int n   = blockIdx.y;
    const int tid = threadIdx.x;
    const int c0  = blockIdx.x * TILE_C;

    __shared__ float  sDelta[36 * TILE_C];   // 36 KB channel-major tile
    __shared__ float4 sg[M_GT];
    __shared__ float  sarea[M_GT];

#if HAVE_TDM
    if (tid < 32) {   // wave 0 issues one TDM op for the whole block tile
        const unsigned long long ga =
            (unsigned long long)(uintptr_t)(deltas + (size_t)n * 36 * HWC + c0);
        const unsigned ldsA =
            (unsigned)(unsigned long long)(AS3 const void*)&sDelta[0];
        const unsigned td0 = (unsigned)(HWC - c0);   // remaining row (OOB -> 0)
        const unsigned td1 = 36u;
        const unsigned long long str0 = (unsigned long long)HWC;

        v4u g0;                                    // D# group 0 (ISA 8.3)
        g0[0] = 1u;                                // count=1, user descriptor
        g0[1] = ldsA;                              // lds_addr
        g0[2] = (unsigned)ga;                      // global_addr[31:0]
        g0[3] = (unsigned)(ga >> 32) | (2u << 30); // addr[56:32] | type=2

        v8i g1;                                    // D# group 1 (ISA 8.4)
        g1[0] = (int)(2u << 16);                   // mask=0, data_size=4B
        g1[1] = (int)((td0 & 0xFFFFu) << 16);      // tensor_dim0[15:0]
        g1[2] = (int)(((td0 >> 16) & 0xFFFFu) | ((td1 & 0xFFFFu) << 16));
        g1[3] = (int)(((td1 >> 16) & 0xFFFFu) | ((unsigned)TILE_C << 16)); // tile_dim0
        g1[4] = (int)36u;                          // tile_dim1=36, tile_dim2=0
        g1[5] = (int)(unsigned)(str0 & 0xFFFFFFFFu);        // dim0_stride lo32
        g1[6] = (int)(unsigned)((str0 >> 32) & 0xFFFFu);    // dim0_stride hi16
        g1[7] = 0;

        v4i g2 = {0, 0, 0, 0};
        v4i g3 = {0, 0, 0, 0};
#if __has_include(<hip/amd_detail/amd_gfx1250_TDM.h>)
        v8i g4 = {0, 0, 0, 0, 0, 0, 0, 0};         // therock 6-arg form
        __builtin_amdgcn_tensor_load_to_lds(g0, g1, g2, g3, g4, 0);
#else
        __builtin_amdgcn_tensor_load_to_lds(g0, g1, g2, g3, 0); // ROCm 7.2 5-arg
#endif
        __builtin_amdgcn_s_wait_tensorcnt(0);
    }
#else
    for (int idx = tid; idx < 36 * TILE_C; idx += 256) {
        const int ch = idx >> 8, i = idx & 255;
        const int cc = c0 + i;
        sDelta[idx] = (cc < HWC) ? deltas[((size_t)n * 36 + ch) * HWC + cc] : 0.0f;
    }
#endif

#if HAVE_ASYNC
    if (tid < M_GT) {   // 64 lanes each DMA one float4 box into LDS
        __builtin_amdgcn_global_load_async_to_lds_b128(
            (gvp4)(gt + ((size_t)n * M_GT + tid) * 4),   // AS1 int32x4*
            (lvp4)&sg[tid],                               // AS3 int32x4*
            0, 0);
    }
    wait_async0();      // issuing waves drain ASYNCcnt before the barrier
#else
    for (int i = tid; i < M_GT; i += blockDim.x)
        sg[i] = ((const float4*)gt)[n * M_GT + i];
#endif
    __syncthreads();

    for (int i = tid; i < M_GT; i += blockDim.x) {
        float4 g = sg[i];
        sarea[i] = fmaxf(g.z - g.x, 0.0f) * fmaxf(g.w - g.y, 0.0f);
    }
    __syncthreads();

    const int c = c0 + tid;             // cell = h*W + w
    if (c < HWC) {
        const int h = c / WW, w = c % WW;
        const float cx = (w + 0.5f) * 16.0f;
        const float cy = (h + 0.5f) * 16.0f;
        int* mout = matches + (size_t)n * A_TOT + (size_t)c * KA;

#pragma unroll
        for (int k = 0; k < KA; ++k) {
            const float dx1 = sDelta[(4 * k + 0) * TILE_C + tid];
            const float dy1 = sDelta[(4 * k + 1) * TILE_C + tid];
            const float dx2 = sDelta[(4 * k + 2) * TILE_C + tid];
            const float dy2 = sDelta[(4 * k + 3) * TILE_C + tid];
            const float ax1 = cx - 0.5f * c_ws[k], ay1 = cy - 0.5f * c_hs[k];
            const float ax2 = cx + 0.5f * c_ws[k], ay2 = cy + 0.5f * c_hs[k];
            const float x1 = clip1920(ax1 + dx1), y1 = clip1920(ay1 + dy1);
            const float x2 = clip1920(ax2 + dx2), y2 = clip1920(ay2 + dy2);
            const float ab = fmaxf(x2 - x1, 0.0f) * fmaxf(y2 - y1, 0.0f);

            float best = -1.0f; int arg = 0;
#pragma unroll 4
            for (int m = 0; m < M_GT; ++m) {
                float4 g = sg[m];
                float ix1 = fmaxf(x1, g.x), iy1 = fmaxf(y1, g.y);
                float ix2 = fminf(x2, g.z), iy2 = fminf(y2, g.w);
                float inter = fmaxf(ix2 - ix1, 0.0f) * fmaxf(iy2 - iy1, 0.0f);
                float iou = inter / (ab + sarea[m] - inter + EPSV);
                if (iou > best) { best = iou; arg = m; }  // strict > == first argmax
            }
            mout[k] = (best >= UPPER_T) ? arg : ((best < LOWER_T) ? -1 : -2);
        }
    }
}

// ---------------------------------------------------------------------------
// Kernel 2: deterministic sampling. One block per image scans anchors in
// index order; block-wide ballot_w32 + prefix scan appends up to 128 pos /
// 60 neg. Raw totals stored; caps applied at loss time (== min(count,cap)).
// ---------------------------------------------------------------------------
__global__ void __launch_bounds__(256)
sample_kernel(const int* __restrict__ matches, int* __restrict__ posIdx,
              int* __restrict__ posGt, int* __restrict__ negIdx,
              int* __restrict__ counts) {
    const int n = blockIdx.x;
    const int tid = threadIdx.x, lane = tid & 31, wv = tid >> 5;
    const int* m = matches + (size_t)n * A_TOT;

    __shared__ int wP[8], wN[8], woP[8], woN[8];
    __shared__ int basePos, baseNeg;
    if (tid == 0) { basePos = 0; baseNeg = 0; }
    __syncthreads();

    for (int chunk = 0; chunk < A_TOT; chunk += 256) {
        const int a  = chunk + tid;
        const int mv = (a < A_TOT) ? m[a] : -2;
        const bool isP = (mv >= 0), isN = (mv == -1);

        const unsigned bp = ballot32(isP), bn = ballot32(isN);
        const unsigned below = (lane == 0) ? 0u : (0xffffffffu >> (32 - lane));
        const int rp = __popc(bp & below), rn = __popc(bn & below);
        if (lane == 0) { wP[wv] = __popc(bp); wN[wv] = __popc(bn); }
        __syncthreads();
        if (tid == 0) {
            int sp = basePos, sn = baseNeg;
#pragma unroll
            for (int i = 0; i < 8; ++i) { woP[i] = sp; sp += wP[i]; woN[i] = sn; sn += wN[i]; }
            basePos = sp; baseNeg = sn;
        }
        __syncthreads();
        if (isP) {
            int s = woP[wv] + rp;
            if (s < NPOSMAX) { posIdx[n * NPOSMAX + s] = a; posGt[n * NPOSMAX + s] = mv; }
        }
        if (isN) {
            int s = woN[wv] + rn;
            if (s < NNEGMAX) negIdx[n * NNEGMAX + s] = a;
        }
        __syncthreads();
    }
    if (tid == 0) { counts[2 * n] = basePos; counts[2 * n + 1] = baseNeg; }
}

// ---------------------------------------------------------------------------
// Kernel 3: per-sample BCE + smooth-L1, reduced via LDS float atomics
// (ds_add_f32). Writes a 16-float stat tile per image for the WMMA reduce:
//   [cls_loss, bbox_loss, nneg, npos, pred_mean, 0...]
// ---------------------------------------------------------------------------
__global__ void __launch_bounds__(256)
loss_kernel(const float* __restrict__ cls, const float* __restrict__ deltas,
            const float* __restrict__ gt, const int* __restrict__ posIdx,
            const int* __restrict__ posGt, const int* __restrict__ negIdx,
            const int* __restrict__ counts, float* __restrict__ stats) {
    const int n = blockIdx.x, tid = threadIdx.x;
    __shared__ float sb[4];   // bce, l1, sum(lp), sum(ln)
    if (tid < 4) sb[tid] = 0.0f;
    __syncthreads();

    const int npos = min(counts[2 * n], NPOSMAX);
    const int nneg = min(counts[2 * n + 1], NNEGMAX);

    float bce = 0.0f, l1 = 0.0f, lps = 0.0f, lns = 0.0f;

    if (tid < npos) {                               // positive slot
        const int a  = posIdx[n * NPOSMAX + tid];
        const int gi = posGt[n * NPOSMAX + tid];
        const int k = a % KA, c = a / KA, h = c / WW, w = c % WW;

        const float lp = cls[((size_t)n * KA + k) * HWC + c];
        bce = softplusf(-lp); lps = lp;

        const float* db = deltas + ((size_t)n * 36 + 4 * k) * HWC + c;
        const float cx = (w + 0.5f) * 16.0f, cy = (h + 0.5f) * 16.0f;
        const float ax1 = cx - 0.5f * c_ws[k], ay1 = cy - 0.5f * c_hs[k];
        const float ax2 = cx + 0.5f * c_ws[k], ay2 = cy + 0.5f * c_hs[k];
        const float x1 = clip1920(ax1 + db[0]),       y1 = clip1920(ay1 + db[HWC]);
        const float x2 = clip1920(ax2 + db[2 * HWC]), y2 = clip1920(ay2 + db[3 * HWC]);

        const float aw = ax2 - ax1, ah = ay2 - ay1;
        const float acx = ax1 + 0.5f * aw, acy = ay1 + 0.5f * ah;
        // encode(region, anchor)
        const float bw = fmaxf(x2 - x1, EPSV), bh = fmaxf(y2 - y1, EPSV);
        const float bcx = x1 + 0.5f * bw, bcy = y1 + 0.5f * bh;
        const float tp0 = (bcx - acx) / aw, tp1 = (bcy - acy) / ah;
        const float tp2 = logf(bw / aw),    tp3 = logf(bh / ah);
        // encode(gt, anchor)
        const float4 g = ((const float4*)gt)[n * M_GT + gi];
        const float gw = fmaxf(g.z - g.x, EPSV), gh = fmaxf(g.w - g.y, EPSV);
        const float gcx = g.x + 0.5f * gw, gcy = g.y + 0.5f * gh;
        const float tg0 = (gcx - acx) / aw, tg1 = (gcy - acy) / ah;
        const float tg2 = logf(gw / aw),    tg3 = logf(gh / ah);

        l1 = smoothl1(tp0 - tg0) + smoothl1(tp1 - tg1) +
             smoothl1(tp2 - tg2) + smoothl1(tp3 - tg3);
    } else if (tid >= NPOSMAX && tid < NPOSMAX + nneg) {  // negative slot
        const int a = negIdx[n * NNEGMAX + tid - NPOSMAX];
        const int k = a % KA, c = a / KA;
        const float ln = cls[((size_t)n * KA + k) * HWC + c];
        bce = softplusf(ln); lns = ln;
    }

    atomicAdd(&sb[0], bce);
    atomicAdd(&sb[1], l1);
    atomicAdd(&sb[2], lps);
    atomicAdd(&sb[3], lns);
    __syncthreads();

    if (tid == 0) {
        const float denom = fmaxf((float)(npos + nneg), 1.0f);
        float* s = stats + n * 16;
        s[0] = sb[0] / denom;                                    // cls_loss
        s[1] = sb[1] / (fmaxf((float)npos, 1.0f) * (float)N_IMG);// bbox_loss
        s[2] = (float)nneg;
        s[3] = (float)npos;
        s[4] = (sb[2] + sb[3]) / denom;                          // pred_mean
#pragma unroll
        for (int j = 5; j < 16; ++j) s[j] = 0.0f;
    }
}

// ---------------------------------------------------------------------------
// Kernel 4: cross-image reduction as D = ones(16x4) x B(4x16) + C with
// V_WMMA_F32_16X16X4_F32 (exact f32). B[k][col] = stats[image][col]; two
// chained WMMAs cover all 8 images. One wave, EXEC all-ones as required.
// Output order: (cls.sum, bbox.sum, nneg.sum, npos.sum, pred_mean[last]).
// ---------------------------------------------------------------------------
__global__ void __launch_bounds__(32)
reduce_kernel(const float* __restrict__ stats, float* __restrict__ out) {
    const int lane = threadIdx.x;
    const int col  = lane & 15;
    const int hi   = (lane >> 4) & 1;          // lanes 16-31 supply K=2,3 rows

    float tot;
#if __has_builtin(__builtin_amdgcn_wmma_f32_16x16x4_f32)
    v8f c = {};
    v2f a; a[0] = 1.0f; a[1] = 1.0f;           // A = ones(16x4), all lanes
#pragma unroll
    for (int i = 0; i < 2; ++i) {
        const int img0 = i * 4 + hi * 2;       // rows K={0,1} or {2,3}
        v2f b;
        b[0] = stats[(img0 + 0) * 16 + col];
        b[1] = stats[(img0 + 1) * 16 + col];
        c = __builtin_amdgcn_wmma_f32_16x16x4_f32(
                false, a, false, b, (short)0, c, false, false);
    }
    tot = c[0];                                 // D[0][col] on lanes 0-15
#else
    tot = 0.0f;
#pragma unroll
    for (int img = 0; img < N_IMG; ++img) tot += stats[img * 16 + col];
#endif

    if (!hi) {
        if (col < 4) out[col] = tot;                    // cl, bl, bg, fg sums
        if (col == 4) out[4] = stats[7 * 16 + 4];       // pred_mean of last image
    }
}

// ---------------------------------------------------------------------------
extern "C" void kernel_launch(void* const* d_in, const int* in_sizes, int n_in,
                              void* d_out, int out_size, void* d_ws, size_t ws_size,
                              hipStream_t stream) {
    const float* cls    = (const float*)d_in[0];  // [8, 9, 120, 120]
    const float* deltas = (const float*)d_in[1];  // [8, 36, 120, 120]
    const float* gt     = (const float*)d_in[2];  // [8, 64, 4]
    float* out = (float*)d_out;

    char* ws = (char*)d_ws;
    size_t off = 0;
    int* matches = (int*)(ws + off); off += (size_t)N_IMG * A_TOT * sizeof(int);
    int* posIdx  = (int*)(ws + off); off += (size_t)N_IMG * NPOSMAX * sizeof(int);
    int* posGt   = (int*)(ws + off); off += (size_t)N_IMG * NPOSMAX * sizeof(int);
    int* negIdx  = (int*)(ws + off); off += (size_t)N_IMG * NNEGMAX * sizeof(int);
    int* counts  = (int*)(ws + off); off += (size_t)N_IMG * 2 * sizeof(int);
    float* stats = (float*)(ws + off);

    match_kernel<<<dim3((HWC + TILE_C - 1) / TILE_C, N_IMG), 256, 0, stream>>>(deltas, gt, matches);
    sample_kernel<<<N_IMG, 256, 0, stream>>>(matches, posIdx, posGt, negIdx, counts);
    loss_kernel<<<N_IMG, 256, 0, stream>>>(cls, deltas, gt, posIdx, posGt, negIdx, counts, stats);
    reduce_kernel<<<1, 32, 0, stream>>>(stats, out);
}